// GRUDecoder_71966472012408
// MI455X (gfx1250) — compile-verified
//
#include <hip/hip_runtime.h>

#define VOCAB 32000
#define EMB   64
#define HID   64
#define BATCH 16
#define SEQ   256
#define NTOK  (BATCH * SEQ)   // 4096
#define G3    (3 * HID)       // 192

typedef __attribute__((ext_vector_type(2))) float v2f;
typedef __attribute__((ext_vector_type(8))) float v8f;

// ---------------------------------------------------------------------------
// Kernel 1: embedding lookup + input projection
//   xp[t, g] = dot(emb[tokens[t]], w_ih[g, :]) + b_ih[g],  t in [0,4096), g in [0,192)
// ---------------------------------------------------------------------------
__global__ __launch_bounds__(G3) void k_embed_xproj(
    const int* __restrict__ tokens, const float* __restrict__ emb,
    const float* __restrict__ w_ih, const float* __restrict__ b_ih,
    float* __restrict__ xp) {
  __shared__ __align__(16) float e[EMB];
  const int t = blockIdx.x;
  const int g = threadIdx.x;
  if (g < EMB) e[g] = emb[(size_t)tokens[t] * EMB + g];
  __syncthreads();
  const float* w = w_ih + (size_t)g * EMB;
  float acc = b_ih[g];
#pragma unroll
  for (int j = 0; j < EMB; j += 4) {
    float4 wv = *(const float4*)(w + j);
    float4 ev = *(const float4*)(e + j);
    acc = fmaf(wv.x, ev.x, acc);
    acc = fmaf(wv.y, ev.y, acc);
    acc = fmaf(wv.z, ev.z, acc);
    acc = fmaf(wv.w, ev.w, acc);
  }
  xp[(size_t)t * G3 + g] = acc;
}

// ---------------------------------------------------------------------------
// Kernel 2: GRU recurrence, one block per batch row (16 blocks x 192 threads).
// Thread g owns gate-output g; its w_hh row lives in VGPRs for all 256 steps.
// h lives in LDS (64 floats), hp staging in LDS (192 floats).
// Gate order (PyTorch): r, z, n.
// ---------------------------------------------------------------------------
__global__ __launch_bounds__(G3) void k_gru(
    const float* __restrict__ xp, const float* __restrict__ w_hh,
    const float* __restrict__ b_hh, float* __restrict__ hs) {
  const int b = blockIdx.x;
  const int g = threadIdx.x;
  __shared__ __align__(16) float hcur[HID];
  __shared__ float hp[G3];

  float wreg[HID];
#pragma unroll
  for (int j = 0; j < HID; j += 4) {
    float4 wv = *(const float4*)(w_hh + (size_t)g * HID + j);
    wreg[j + 0] = wv.x; wreg[j + 1] = wv.y;
    wreg[j + 2] = wv.z; wreg[j + 3] = wv.w;
  }
  const float bh = b_hh[g];
  if (g < HID) hcur[g] = 0.0f;
  __syncthreads();

  const float* xrow = xp + (size_t)b * SEQ * G3;
  float* hrow = hs + (size_t)b * SEQ * HID;

  for (int s = 0; s < SEQ; ++s) {
    // hp[g] = b_hh[g] + w_hh[g,:] . h
    float acc = bh;
#pragma unroll
    for (int j = 0; j < HID; j += 4) {
      float4 hv = *(const float4*)(hcur + j);   // LDS broadcast reads
      acc = fmaf(wreg[j + 0], hv.x, acc);
      acc = fmaf(wreg[j + 1], hv.y, acc);
      acc = fmaf(wreg[j + 2], hv.z, acc);
      acc = fmaf(wreg[j + 3], hv.w, acc);
    }
    hp[g] = acc;
    __syncthreads();
    if (g < HID) {
      const float* xt = xrow + (size_t)s * G3;
      float r = 1.0f / (1.0f + __expf(-(xt[g] + hp[g])));
      float z = 1.0f / (1.0f + __expf(-(xt[HID + g] + hp[HID + g])));
      float n = tanhf(xt[2 * HID + g] + r * hp[2 * HID + g]);
      float hn = (1.0f - z) * n + z * hcur[g];
      hcur[g] = hn;
      hrow[(size_t)s * HID + g] = hn;
    }
    __syncthreads();
  }
}

// ---------------------------------------------------------------------------
// Kernel 3: logits = hs[4096,64] @ w_out^T[64,32000] + b_out  via
// V_WMMA_F32_16X16X4_F32 (full fp32). One wave computes a 16x64 output strip
// (4 N-tiles), amortizing its A fragments. Block = 4 waves -> 16x256 of out.
// Grid = (32000/256, 4096/16) = (125, 256).
// Store-bandwidth bound (524 MB out) -> nontemporal stores to spare L2.
// ---------------------------------------------------------------------------
__global__ __launch_bounds__(128) void k_logits_wmma(
    const float* __restrict__ hsm, const float* __restrict__ w_out,
    const float* __restrict__ b_out, float* __restrict__ out) {
  const int lane = threadIdx.x & 31;
  const int wid  = threadIdx.x >> 5;
  const int m0    = blockIdx.y * 16;
  const int nbase = blockIdx.x * 256 + wid * 64;

  // ISA 32-bit A 16x4 layout: lane&15 = M row; lanes0-15 hold K{0,1},
  // lanes16-31 hold K{2,3} within the 2-VGPR fragment.
  const int row  = lane & 15;
  const int koff = (lane >> 4) * 2;
  const int col  = lane & 15;   // B/C/D: lane&15 = N column

  v2f afrag[16];
  const float* Arow = hsm + (size_t)(m0 + row) * HID + koff;
#pragma unroll
  for (int kk = 0; kk < 16; ++kk)
    afrag[kk] = *(const v2f*)(Arow + 4 * kk);

#pragma unroll
  for (int nt = 0; nt < 4; ++nt) {
    const int n0 = nbase + nt * 16;
    const float bias = b_out[n0 + col];
    v8f acc = {bias, bias, bias, bias, bias, bias, bias, bias};
    // B[k][n] = w_out[n][k]; same K-split across lane halves as A.
    const float* Bcol = w_out + (size_t)(n0 + col) * HID + koff;
#pragma unroll
    for (int kk = 0; kk < 16; ++kk) {
      v2f bfrag = *(const v2f*)(Bcol + 4 * kk);
      acc = __builtin_amdgcn_wmma_f32_16x16x4_f32(
          false, afrag[kk], false, bfrag, (short)0, acc, false, false);
    }
    const int r8 = (lane >> 4) * 8;  // C/D: VGPR v -> row m0+v (lo) / m0+v+8 (hi)
#pragma unroll
    for (int v = 0; v < 8; ++v) {
      __builtin_nontemporal_store(
          acc[v], &out[(size_t)(m0 + v + r8) * VOCAB + (n0 + col)]);
    }
  }
}

// ---------------------------------------------------------------------------
extern "C" void kernel_launch(void* const* d_in, const int* in_sizes, int n_in,
                              void* d_out, int out_size, void* d_ws, size_t ws_size,
                              hipStream_t stream) {
  const int*   tokens = (const int*)d_in[0];
  const float* emb    = (const float*)d_in[1];
  const float* w_ih   = (const float*)d_in[2];
  const float* w_hh   = (const float*)d_in[3];
  const float* b_ih   = (const float*)d_in[4];
  const float* b_hh   = (const float*)d_in[5];
  const float* w_out  = (const float*)d_in[6];
  const float* b_out  = (const float*)d_in[7];
  float* out = (float*)d_out;

  float* xp = (float*)d_ws;               // 4096 * 192 floats = 3 MB
  float* hs = xp + (size_t)NTOK * G3;     // 4096 * 64 floats  = 1 MB

  k_embed_xproj<<<NTOK, G3, 0, stream>>>(tokens, emb, w_ih, b_ih, xp);
  k_gru<<<BATCH, G3, 0, stream>>>(xp, w_hh, b_hh, hs);

  dim3 grid3(VOCAB / 256, NTOK / 16, 1);  // (125, 256)
  k_logits_wmma<<<grid3, 128, 0, stream>>>(hs, w_out, b_out, out);
}